// Choquet_integral_3289944949020
// MI455X (gfx1250) — compile-verified
//
#include <hip/hip_runtime.h>
#include <stdint.h>

#define N_IN    12
#define N_OUT   16
#define N_VARS  4094                     // 2^12 - 2
#define FM_ROWS (N_VARS + 1)             // 4095 (ones row appended)
#define FM_ELEMS (FM_ROWS * N_OUT)       // 65520 floats
#define FM_BYTES (FM_ELEMS * 4)          // 262,080 B
#define BLOCK    256
#define FI_BYTES (BLOCK * N_IN * 2)      // 6,144 B  (u16 FM indices)
#define D_BYTES  (BLOCK * N_IN * 4)      // 12,288 B (f32 diffs)
#define SMEM2    (FM_BYTES + FI_BYTES + D_BYTES)   // 280,512 B <= 320KB WGP LDS

typedef __attribute__((ext_vector_type(8))) float v8f;
typedef __attribute__((ext_vector_type(2))) float v2f;

// ---------------------------------------------------------------------------
// Kernel 1: build the fuzzy-measure lattice in LDS (262KB fits CDNA5's 320KB
// WGP LDS), one workgroup, level-by-level with barriers, write clamped FM.
// ---------------------------------------------------------------------------
__global__ void __launch_bounds__(1024)
build_fm_kernel(const float* __restrict__ fm_vars, float* __restrict__ fm_out) {
    extern __shared__ float fm[];            // N_VARS * N_OUT floats
    const int tid = threadIdx.x;
    const int nt  = blockDim.x;

    for (int t = tid; t < N_IN * N_OUT; t += nt) {   // singletons
        int i = t >> 4, o = t & 15;
        int r = (1 << i) - 1;
        fm[r * N_OUT + o] = fabsf(fm_vars[r * N_OUT + o]);
    }
    __syncthreads();

    for (int k = 2; k < N_IN; ++k) {                 // levels 2..11
        for (int t = tid; t < N_VARS * N_OUT; t += nt) {
            int s = (t >> 4) + 1;
            if (__popc(s) != k) continue;
            int o = t & 15;
            float mx = 0.0f;
            #pragma unroll
            for (int b = 0; b < N_IN; ++b)
                if (s & (1 << b))
                    mx = fmaxf(mx, fm[((s - (1 << b)) - 1) * N_OUT + o]);
            fm[(s - 1) * N_OUT + o] = mx + fabsf(fm_vars[(s - 1) * N_OUT + o]);
        }
        __syncthreads();
    }

    for (int t = tid; t < FM_ELEMS; t += nt) {
        float v = (t >= N_VARS * N_OUT) ? 1.0f : fminf(fm[t], 1.0f);
        fm_out[t] = v;
    }
}

// ---------------------------------------------------------------------------
// Kernel 2: Choquet integral.
//  1) async-stage the 262KB FM table to LDS (ASYNCcnt path, CDNA5)
//  2) per-lane in-register sort of 12 inputs (overlapped with the copy)
//  3) out = A @ B via V_WMMA_F32_16X16X4_F32: A = block-diagonal diffs
//     (16x192), B = gathered FM rows (192x16) read from LDS.
// ---------------------------------------------------------------------------
__global__ void __launch_bounds__(BLOCK)
choquet_wmma_kernel(const float* __restrict__ inputs,
                    const float* __restrict__ fm_g,
                    float* __restrict__ out, int M) {
    extern __shared__ char smem[];
    float*          fm    = (float*)smem;                       // [FM_ELEMS]
    unsigned short* lfi   = (unsigned short*)(smem + FM_BYTES); // [BLOCK*12]
    float*          ldsd  = (float*)(smem + FM_BYTES + FI_BYTES); // [BLOCK*12]

    const int tid = threadIdx.x;

    // ---- CDNA5 async global -> LDS staging of the FM table
    {
        uint32_t lds_base = (uint32_t)(uintptr_t)&fm[0];
        const int nchunks = FM_BYTES / 16;                      // 16380 x b128
        for (int c = tid; c < nchunks; c += BLOCK) {
            uint32_t off = (uint32_t)c * 16u;
            asm volatile("global_load_async_to_lds_b128 %0, %1, %2"
                         :: "v"(lds_base + off), "v"(off), "s"(fm_g)
                         : "memory");
        }
    }

    const int row = blockIdx.x * BLOCK + tid;

    // ---- load + sort own row while the async copy is in flight
    float          dvals[N_IN];
    unsigned short fidx[N_IN];
    if (row < M) {
        const float* ip = inputs + row * N_IN;                  // 48B, 16B-aligned
        float4 a = *(const float4*)(ip + 0);
        float4 b = *(const float4*)(ip + 4);
        float4 c = *(const float4*)(ip + 8);
        float v[N_IN] = {a.x, a.y, a.z, a.w, b.x, b.y, b.z, b.w,
                         c.x, c.y, c.z, c.w};
        uint64_t key[N_IN];
        #pragma unroll
        for (int j = 0; j < N_IN; ++j)
            key[j] = ((uint64_t)__float_as_uint(v[j]) << 8) |
                     (uint64_t)(N_IN - 1 - j);                  // stable-desc keys
        #pragma unroll
        for (int p = 0; p < N_IN; ++p) {                        // odd-even sort
            #pragma unroll
            for (int i = (p & 1); i + 1 < N_IN; i += 2) {
                uint64_t lo = key[i], hi = key[i + 1];
                bool sw = lo < hi;
                key[i]     = sw ? hi : lo;
                key[i + 1] = sw ? lo : hi;
            }
        }
        uint32_t mask = 0u;
        #pragma unroll
        for (int j = 0; j < N_IN; ++j) {
            float sj = __uint_as_float((uint32_t)(key[j] >> 8));
            float sn = (j + 1 < N_IN)
                         ? __uint_as_float((uint32_t)(key[j + 1] >> 8)) : 0.0f;
            mask |= 1u << ((uint32_t)(N_IN - 1) - (uint32_t)(key[j] & 0xFFu));
            dvals[j] = sj - sn;
            fidx[j]  = (unsigned short)(mask - 1u);             // FM row index
        }
    } else {
        #pragma unroll
        for (int j = 0; j < N_IN; ++j) { dvals[j] = 0.0f; fidx[j] = 0; }
    }
    #pragma unroll
    for (int j = 0; j < N_IN; ++j) {                            // publish to LDS
        lfi[tid * N_IN + j]  = fidx[j];
        ldsd[tid * N_IN + j] = dvals[j];
    }

    asm volatile("s_wait_asynccnt 0x0" ::: "memory");
    __syncthreads();

    // ---- WMMA accumulation: per wave, two 16-row tiles (rows = its 32 lanes)
    const int lane     = tid & 31;
    const int waveBase = tid & ~31;          // first row (block-local) of wave
    const int hi       = lane >> 4;          // K half-select (A/B VGPR layout)
    const int nsel     = lane & 15;          // M (A) / N (B,C/D) lane index

    #pragma unroll
    for (int t = 0; t < 2; ++t) {
        const int tileBase = waveBase + t * 16;
        v8f acc = {};                        // C = 0
        for (int mp = 0; mp < 16; ++mp) {    // diagonal block (tile row) m'
            const int fb = (tileBase + mp) * N_IN;
            #pragma unroll
            for (int u = 0; u < 3; ++u) {    // 3 K-chunks of 4 per row block
                const int j0 = 4 * u + 2 * hi;
                int   i0 = lfi[fb + j0];
                int   i1 = lfi[fb + j0 + 1];
                float b0 = fm[i0 * N_OUT + nsel];
                float b1 = fm[i1 * N_OUT + nsel];
                float d0 = ldsd[fb + j0];
                float d1 = ldsd[fb + j0 + 1];
                v2f A, B;
                A.x = (nsel == mp) ? d0 : 0.0f;   // block-diagonal diffs
                A.y = (nsel == mp) ? d1 : 0.0f;
                B.x = b0;                         // gathered FM rows
                B.y = b1;
                acc = __builtin_amdgcn_wmma_f32_16x16x4_f32(
                        /*neg_a=*/false, A, /*neg_b=*/false, B,
                        /*c_mod=*/(short)0, acc,
                        /*reuse_a=*/false, /*reuse_b=*/false);
            }
        }
        // ---- store C (16x16 f32 layout: VGPR r -> M=r / M=r+8 per lane half)
        const int gRow0 = blockIdx.x * BLOCK + tileBase + 8 * hi;
        float* op = out + (size_t)gRow0 * N_OUT + nsel;
        #pragma unroll
        for (int r = 0; r < 8; ++r)
            if (gRow0 + r < M) op[r * N_OUT] = acc[r];
    }
}

// ---------------------------------------------------------------------------
extern "C" void kernel_launch(void* const* d_in, const int* in_sizes, int n_in,
                              void* d_out, int out_size, void* d_ws, size_t ws_size,
                              hipStream_t stream) {
    const float* inputs  = (const float*)d_in[0];   // [M, 12] fp32
    const float* fm_vars = (const float*)d_in[1];   // [4094, 16] fp32
    float* outp = (float*)d_out;                    // [M, 16] fp32
    float* fm   = (float*)d_ws;                     // [4095, 16] fp32 scratch

    const int M = in_sizes[0] / N_IN;               // 16384

    build_fm_kernel<<<1, 1024, (size_t)N_VARS * N_OUT * sizeof(float), stream>>>(
        fm_vars, fm);

    const int grid = (M + BLOCK - 1) / BLOCK;       // 64 blocks
    choquet_wmma_kernel<<<grid, BLOCK, (size_t)SMEM2, stream>>>(
        inputs, fm, outp, M);
}